// PNA_18073222382225
// MI455X (gfx1250) — compile-verified
//
#include <hip/hip_runtime.h>
#include <math.h>

typedef __attribute__((ext_vector_type(2))) float v2f;
typedef __attribute__((ext_vector_type(8))) float v8f;

#define F_IN   40
#define TOW    5
#define COLS   200      // T * F_IN message channels
#define COLS_PAD 208    // padded to 13 tiles of 16
#define NT     13       // n-tiles
#define KDIM   80       // 2*F_IN
#define KSTEPS 20       // KDIM/4
#define ET     4        // edge-tiles per block (64 edges)
#define F_T    8
#define EMB    40
#define OUTD   2
#define EPS_STD 1e-5f
#define EPS_GN  1e-5f

// ---------------- CSR build ----------------

__global__ void k_hist(const int* __restrict__ dst, int* __restrict__ cnt, int E) {
    int e = blockIdx.x * blockDim.x + threadIdx.x;
    if (e < E) atomicAdd(&cnt[dst[e]], 1);
}

__global__ void k_scan(const int* __restrict__ cnt, int* __restrict__ rowp, int N) {
    __shared__ int part[1024];
    int t = threadIdx.x;
    int CH = (N + 1023) >> 10;
    int s0 = t * CH, s1 = min(s0 + CH, N);
    int s = 0;
    for (int i = s0; i < s1; ++i) s += cnt[i];
    part[t] = s;
    __syncthreads();
    for (int off = 1; off < 1024; off <<= 1) {
        int v = (t >= off) ? part[t - off] : 0;
        __syncthreads();
        part[t] += v;
        __syncthreads();
    }
    int run = (t == 0) ? 0 : part[t - 1];
    for (int i = s0; i < s1; ++i) { rowp[i] = run; run += cnt[i]; }
    if (s0 < N && s1 == N) rowp[N] = run;
}

__global__ void k_scatter(const int* __restrict__ dst, const int* __restrict__ rowp,
                          int* __restrict__ fill, int* __restrict__ eord, int E) {
    int e = blockIdx.x * blockDim.x + threadIdx.x;
    if (e < E) {
        int d = dst[e];
        int pos = rowp[d] + atomicAdd(&fill[d], 1);
        eord[pos] = e;
    }
}

// ---------------- weight transpose: Wt[n][k] = W_pre[n/40][k][n%40] ----------------

__global__ void k_prep_wt(const float* __restrict__ Wpre, float* __restrict__ Wt) {
    int idx = blockIdx.x * blockDim.x + threadIdx.x;
    if (idx >= COLS_PAD * KDIM) return;
    int n = idx / KDIM, k = idx % KDIM;
    float v = 0.f;
    if (n < COLS) {
        int t = n / F_IN, g = n % F_IN;
        v = Wpre[((size_t)t * KDIM + k) * F_IN + g];
    }
    Wt[idx] = v;
}

// ---------------- edge message GEMM via V_WMMA_F32_16X16X4_F32 ----------------
// Block = 128 threads, stages 64 edges (4 edge-tiles) of A in LDS.
// Each wave owns a disjoint chunk of n-tiles (4/3/3/3) and computes
// 4 edge-tiles x NTC n-tiles per k-step, so every A/B fragment load
// feeds multiple WMMAs (16 WMMAs per 8 loads per wave).

template <int NTC>
__device__ __forceinline__ void gemm_chunk(
        const float* __restrict__ sH,        // [ET][16][KDIM] LDS
        const float* __restrict__ Wt,        // [COLS_PAD][KDIM] global (L2-hot)
        const float* __restrict__ bpre,
        float* __restrict__ m,
        int ntStart, int ebase, int E, int lane) {
    const int half = lane >> 4;
    const int l16  = lane & 15;
    const int aRow = l16 * KDIM;

    v8f acc[ET][NTC];
#pragma unroll
    for (int t = 0; t < ET; ++t)
#pragma unroll
        for (int i = 0; i < NTC; ++i) acc[t][i] = (v8f)(0.f);

    for (int kk = 0; kk < KSTEPS; ++kk) {
        const int kb = 4 * kk + 2 * half;
        v2f a[ET];
#pragma unroll
        for (int t = 0; t < ET; ++t)
            a[t] = *(const v2f*)&sH[t * 16 * KDIM + aRow + kb];
#pragma unroll
        for (int i = 0; i < NTC; ++i) {
            int n = (ntStart + i) * 16 + l16;
            v2f b = *(const v2f*)&Wt[n * KDIM + kb];
#pragma unroll
            for (int t = 0; t < ET; ++t)
                acc[t][i] = __builtin_amdgcn_wmma_f32_16x16x4_f32(
                    false, a[t], false, b, (short)0, acc[t][i], false, false);
        }
    }

    // D layout: VGPR p -> rows p (lanes 0-15) / p+8 (lanes 16-31), col = l16
#pragma unroll
    for (int i = 0; i < NTC; ++i) {
        int n = (ntStart + i) * 16 + l16;
        if (n >= COLS) continue;
        float bias = bpre[n];
#pragma unroll
        for (int t = 0; t < ET; ++t) {
#pragma unroll
            for (int p = 0; p < 8; ++p) {
                int e = ebase + t * 16 + p + 8 * half;
                if (e < E)
                    m[(unsigned)e * COLS + n] = acc[t][i][p] + bias;
            }
        }
    }
}

__global__ __launch_bounds__(128) void k_edge_mlp(
        const float* __restrict__ x, const int* __restrict__ src,
        const int* __restrict__ dst, const float* __restrict__ Wt,
        const float* __restrict__ bpre, float* __restrict__ m, int E) {
    __shared__ __align__(16) float sH[ET * 16 * KDIM];   // 64 edges x 80 feats

    const int wave = threadIdx.x >> 5;
    const int lane = threadIdx.x & 31;
    const int ebase = blockIdx.x * (ET * 16);

    // stage 64 edges x 80 features: [xi(dst) | xj(src)]
    for (int idx = threadIdx.x; idx < ET * 16 * KDIM; idx += 128) {
        int mrow = idx / KDIM, f = idx % KDIM;
        int e = ebase + mrow;
        float v = 0.f;
        if (e < E) {
            int node = (f < F_IN) ? dst[e] : src[e];
            int ff   = (f < F_IN) ? f : f - F_IN;
            v = x[node * F_IN + ff];
        }
        sH[idx] = v;
    }
    __syncthreads();

    if (wave == 0)
        gemm_chunk<4>(sH, Wt, bpre, m, 0, ebase, E, lane);
    else
        gemm_chunk<3>(sH, Wt, bpre, m, 4 + 3 * (wave - 1), ebase, E, lane);
}

// ---------------- per-node reduction + post-MLP + W_lin ----------------

__global__ __launch_bounds__(256) void k_node(
        const float* __restrict__ x, const float* __restrict__ m,
        const int* __restrict__ rowp, const int* __restrict__ eord,
        const float* __restrict__ Wpost, const float* __restrict__ bpost,
        const float* __restrict__ Wlin, const float* __restrict__ blin,
        const float* __restrict__ avg_deg_log,
        float* __restrict__ h, float* __restrict__ gnsum, int N) {
    __shared__ float smMean[COLS], smMin[COLS], smMax[COLS], smStd[COLS];
    __shared__ float smX[F_IN], smTow[EMB];

    const int node = blockIdx.x;
    const int j = threadIdx.x;
    if (node >= N) return;
    const int r0 = rowp[node], r1 = rowp[node + 1];
    const int deg = r1 - r0;

    if (j < F_IN) smX[j] = x[node * F_IN + j];
    if (j < COLS) {
        float s = 0.f, s2 = 0.f, mn = 3.4e38f, mx = -3.4e38f;
        for (int r = r0; r < r1; ++r) {
            int e = eord[r];
            float v = m[(unsigned)e * COLS + j];
            s += v; s2 += v * v;
            mn = fminf(mn, v); mx = fmaxf(mx, v);
        }
        float denom = (float)max(deg, 1);
        float mean = s / denom, mean2 = s2 / denom;
        smMean[j] = mean;
        smStd[j]  = sqrtf(fmaxf(mean2 - mean * mean, 0.f) + EPS_STD);
        smMin[j]  = (deg > 0) ? mn : 0.f;
        smMax[j]  = (deg > 0) ? mx : 0.f;
    }
    __syncthreads();

    if (j < EMB) {  // one thread per (tower, g) output
        const float adl  = avg_deg_log[0];
        const float logd = logf((float)max(deg, 1) + 1.f);
        const float alpha = logd / adl, beta = adl / logd;
        const int tw = j / F_T, g = j % F_T;
        const float* Wp = Wpost + (size_t)tw * 520 * F_T;
        float acc = bpost[tw * F_T + g];
        for (int f = 0; f < F_IN; ++f)            // residual x block (rows 0..39)
            acc += smX[f] * Wp[f * F_T + g];
        for (int f = 0; f < 160; ++f) {           // agg, agg*alpha, agg*beta
            int cat = f / F_IN, ff = f % F_IN;
            int col = tw * F_IN + ff;
            float av = (cat == 0) ? smMean[col] : (cat == 1) ? smMin[col]
                     : (cat == 2) ? smMax[col]  : smStd[col];
            acc += av * (Wp[(40 + f) * F_T + g]
                 + alpha * Wp[(200 + f) * F_T + g]
                 + beta  * Wp[(360 + f) * F_T + g]);
        }
        smTow[j] = acc;
    }
    __syncthreads();

    if (j < EMB) {
        float hv = blin[j];
        for (int c = 0; c < EMB; ++c) hv += smTow[c] * Wlin[c * EMB + j];
        h[node * EMB + j] = hv;
        atomicAdd(&gnsum[j], hv);
    }
}

// ---------------- GraphNorm variance pass ----------------

__global__ void k_var(const float* __restrict__ h, const float* __restrict__ gnsum,
                      const float* __restrict__ ms, float* __restrict__ gnvar,
                      int N) {
    int idx = blockIdx.x * blockDim.x + threadIdx.x;
    if (idx >= N * EMB) return;
    int c = idx % EMB;
    float mean = gnsum[c] / (float)N;
    float d = h[idx] - ms[c] * mean;
    atomicAdd(&gnvar[c], d * d);
}

// ---------------- GraphNorm finalize + ReLU + MLP head ----------------

__global__ __launch_bounds__(128) void k_head(
        const float* __restrict__ h, const float* __restrict__ gnsum,
        const float* __restrict__ gnvar, const float* __restrict__ ms,
        const float* __restrict__ gw, const float* __restrict__ gb,
        const float* __restrict__ W1, const float* __restrict__ b1,
        const float* __restrict__ W2, const float* __restrict__ b2,
        float* __restrict__ out, int N) {
    __shared__ float sW1[EMB * EMB], sW2[EMB * OUTD], sb1[EMB];
    __shared__ float sScale[EMB], sShiftM[EMB], sGb[EMB];
    for (int idx = threadIdx.x; idx < EMB * EMB; idx += blockDim.x) sW1[idx] = W1[idx];
    for (int idx = threadIdx.x; idx < EMB * OUTD; idx += blockDim.x) sW2[idx] = W2[idx];
    if (threadIdx.x < EMB) {
        int c = threadIdx.x;
        sb1[c] = b1[c];
        float mean = gnsum[c] / (float)N;
        sShiftM[c] = ms[c] * mean;
        sScale[c]  = gw[c] * rsqrtf(gnvar[c] / (float)N + EPS_GN);
        sGb[c]     = gb[c];
    }
    __syncthreads();

    int i = blockIdx.x * blockDim.x + threadIdx.x;
    if (i >= N) return;
    float y[EMB];
    for (int c = 0; c < EMB; ++c) {
        float v = sScale[c] * (h[i * EMB + c] - sShiftM[c]) + sGb[c];
        y[c] = fmaxf(v, 0.f);
    }
    float o0 = b2[0], o1 = b2[1];
    for (int jj = 0; jj < EMB; ++jj) {
        float hid = sb1[jj];
        for (int c = 0; c < EMB; ++c) hid += y[c] * sW1[c * EMB + jj];
        hid = fmaxf(hid, 0.f);
        o0 += hid * sW2[jj * OUTD + 0];
        o1 += hid * sW2[jj * OUTD + 1];
    }
    out[i * OUTD + 0] = o0;
    out[i * OUTD + 1] = o1;
}

// ---------------- launch ----------------

extern "C" void kernel_launch(void* const* d_in, const int* in_sizes, int n_in,
                              void* d_out, int out_size, void* d_ws, size_t ws_size,
                              hipStream_t stream) {
    const float* x     = (const float*)d_in[0];
    const int*   ei    = (const int*)d_in[1];
    const float* Wpre  = (const float*)d_in[2];
    const float* bpre  = (const float*)d_in[3];
    const float* Wpost = (const float*)d_in[4];
    const float* bpost = (const float*)d_in[5];
    const float* Wlin  = (const float*)d_in[6];
    const float* blin  = (const float*)d_in[7];
    const float* gw    = (const float*)d_in[8];
    const float* gb    = (const float*)d_in[9];
    const float* gms   = (const float*)d_in[10];
    const float* W1    = (const float*)d_in[11];
    const float* b1    = (const float*)d_in[12];
    const float* W2    = (const float*)d_in[13];
    const float* b2    = (const float*)d_in[14];
    const float* adl   = (const float*)d_in[15];
    float* out = (float*)d_out;

    const int N = in_sizes[0] / F_IN;
    const int E = in_sizes[1] / 2;
    const int* src = ei;
    const int* dst = ei + E;

    char* ws = (char*)d_ws;
    size_t off = 0;
    auto take = [&](size_t bytes) -> char* {
        char* r = ws + off;
        off = (off + bytes + 255) & ~(size_t)255;
        return r;
    };
    float* m    = (float*)take((size_t)E * COLS * sizeof(float));   // 256 MB
    float* Wt   = (float*)take((size_t)COLS_PAD * KDIM * sizeof(float));
    int*   rowp = (int*)take((size_t)(N + 1) * sizeof(int));
    size_t zbytes = ((size_t)2 * N + 128) * sizeof(int);
    char*  zreg = take(zbytes);
    int*   cnt  = (int*)zreg;
    int*   fill = cnt + N;
    float* gnsum = (float*)(fill + N);
    float* gnvar = gnsum + 64;
    int*   eord = (int*)take((size_t)E * sizeof(int));
    float* h    = (float*)take((size_t)N * EMB * sizeof(float));

    hipMemsetAsync(zreg, 0, zbytes, stream);

    k_prep_wt<<<(COLS_PAD * KDIM + 255) / 256, 256, 0, stream>>>(Wpre, Wt);
    k_hist<<<(E + 255) / 256, 256, 0, stream>>>(dst, cnt, E);
    k_scan<<<1, 1024, 0, stream>>>(cnt, rowp, N);
    k_scatter<<<(E + 255) / 256, 256, 0, stream>>>(dst, rowp, fill, eord, E);

    int nblk = (E + ET * 16 - 1) / (ET * 16);
    k_edge_mlp<<<nblk, 128, 0, stream>>>(x, src, dst, Wt, bpre, m, E);

    k_node<<<N, 256, 0, stream>>>(x, m, rowp, eord, Wpost, bpost, Wlin, blin,
                                  adl, h, gnsum, N);
    k_var<<<(N * EMB + 255) / 256, 256, 0, stream>>>(h, gnsum, gms, gnvar, N);
    k_head<<<(N + 127) / 128, 128, 0, stream>>>(h, gnsum, gnvar, gms, gw, gb,
                                                W1, b1, W2, b2, out, N);
}